// GP_70489003262425
// MI455X (gfx1250) — compile-verified
//
#include <hip/hip_runtime.h>
#include <math.h>

typedef __attribute__((ext_vector_type(2))) float v2f;
typedef __attribute__((ext_vector_type(8))) float v8f;

#define HN   16      // neighbors
#define LB   16      // output channels (batch of kernels)
#define DD   3       // spatial dim
#define NPTS 2048
#define WAVES_PER_BLOCK 2
#define JITTER 1e-6f

// Per-wave LDS scratch:
//  d2   : 16x16 pairwise squared distances among neighbors
//  d2nx : 16 squared distances neighbor -> test point
//  diag : squared row norms (Gram diagonal)
//  pl   : per-lane (per-l) private region: chol matrix [16][17] (=272),
//         kx @272 (16), y @288 (16)  -> pad to 320
struct WaveScratch {
    float d2[HN * HN];
    float d2nx[HN];
    float diag[HN];
    float pl[LB][320];
};

__global__ void __launch_bounds__(WAVES_PER_BLOCK * 32)
gp_posterior_kernel(const float* __restrict__ x,        // [N,3]
                    const float* __restrict__ X_near,   // [N,16,3]
                    const float* __restrict__ Z_near,   // [N,16,16]
                    const float* __restrict__ std_Z,    // [N,16,16]
                    const float* __restrict__ lengthscale, // [16]
                    const float* __restrict__ outputscale, // [16]
                    const float* __restrict__ mean_const,  // [16]
                    float* __restrict__ out)            // [2,N,16] (mean, cov)
{
    __shared__ WaveScratch ws[WAVES_PER_BLOCK];

    const int lane = threadIdx.x & 31;
    const int wv   = threadIdx.x >> 5;
    const int n    = blockIdx.x * WAVES_PER_BLOCK + wv;

    WaveScratch& S = ws[wv];

    // ---------------- cooperative phase (full wave, EXEC all 1s) -------------
    const int m    = lane & 15;   // neighbor row this lane mirrors
    const int half = lane >> 4;   // lane half selects K-slice of A operand

    const float* Xn = X_near + (size_t)n * HN * DD;
    const float nx0 = Xn[m * 3 + 0];
    const float nx1 = Xn[m * 3 + 1];
    const float nx2 = Xn[m * 3 + 2];

    const float tx0 = x[n * 3 + 0];
    const float tx1 = x[n * 3 + 1];
    const float tx2 = x[n * 3 + 2];

    // A operand for V_WMMA_F32_16X16X4_F32 (16x4, K=3 zero-padded to 4):
    // VGPR0: lanes0-15 K=0 / lanes16-31 K=2 ; VGPR1: K=1 / K=3(=0)
    v2f a;
    a.x = half ? nx2 : nx0;
    a.y = half ? 0.0f : nx1;

    v8f cz = {};
    // Gram matrix G = Xn * Xn^T  (B operand layout mirrors A; G symmetric)
    v8f g = __builtin_amdgcn_wmma_f32_16x16x4_f32(
        false, a, false, a, (short)0, cz, false, false);

    // Gram diagonal = squared row norm; each lane owns row m directly.
    const float norm = nx0 * nx0 + nx1 * nx1 + nx2 * nx2;
    if (half == 0) S.diag[m] = norm;            // single conditional store

    // cross distances neighbor -> test point (independent, fills the gap)
    {
        float dx0 = nx0 - tx0, dx1 = nx1 - tx1, dx2 = nx2 - tx2;
        if (half == 0) S.d2nx[m] = dx0 * dx0 + dx1 * dx1 + dx2 * dx2;
    }
    __builtin_amdgcn_wave_barrier();            // DS in-order within a wave

    // d2[r][m] = Grr + Gmm - 2 Grm ; Gmm == norm for this lane.
    {
        const float* dp = S.diag + 8 * half;    // this half's 8 row diagonals
#pragma unroll
        for (int k = 0; k < 8; ++k) {
            int r = k + 8 * half;
            float d2 = dp[k] + norm - 2.0f * g[k];
            S.d2[r * HN + m] = fmaxf(d2, 0.0f);
        }
    }
    __builtin_amdgcn_wave_barrier();

    // ---------------- per-channel phase: lane l owns channel l ---------------
    if (lane < LB) {
        const int l = lane;
        const float ls  = lengthscale[l];
        const float ils = 1.0f / (ls * ls);
        const float os  = outputscale[l];
        const float mc  = mean_const[l];

        float* M  = S.pl[l];        // chol matrix, stride 17 (lower triangle)
        float* kx = M + 272;        // K_nx vector
        float* yv = M + 288;        // solve vector

        // build K (lower triangle) and kx
        for (int i = 0; i < HN; ++i) {
            for (int j = 0; j < i; ++j) {
                float d2 = S.d2[i * HN + j];
                M[i * 17 + j] = os * __expf(-0.5f * d2 * ils);
            }
            M[i * 17 + i] = os + JITTER;                 // exp(0)*os + jitter
            kx[i] = os * __expf(-0.5f * S.d2nx[i] * ils);
        }

        // in-place Cholesky (lower)
        for (int i = 0; i < HN; ++i) {
            for (int j = 0; j <= i; ++j) {
                float s = M[i * 17 + j];
                for (int k = 0; k < j; ++k)
                    s -= M[i * 17 + k] * M[j * 17 + k];
                if (j == i) M[i * 17 + i] = sqrtf(fmaxf(s, 1e-20f));
                else        M[i * 17 + j] = s / M[j * 17 + j];
            }
        }

        // forward solve L y = kx
        for (int i = 0; i < HN; ++i) {
            float s = kx[i];
            for (int k = 0; k < i; ++k)
                s -= M[i * 17 + k] * yv[k];
            yv[i] = s / M[i * 17 + i];
        }
        // backward solve L^T a = y (in place)
        for (int i = HN - 1; i >= 0; --i) {
            float s = yv[i];
            for (int k = i + 1; k < HN; ++k)
                s -= M[k * 17 + i] * yv[k];
            yv[i] = s / M[i * 17 + i];
        }

        // reductions + outputs
        const float* Zn = Z_near + (size_t)n * HN * LB + l;
        const float* Sn = std_Z  + (size_t)n * HN * LB + l;
        float pm = 0.0f, cv = 0.0f, ss = 0.0f;
        for (int i = 0; i < HN; ++i) {
            float ai = yv[i];
            pm += ai * (Zn[i * LB] - mc);
            cv += kx[i] * ai;
            float si = Sn[i * LB] * ai;
            ss += si * si;
        }
        out[(size_t)n * LB + l]                       = pm + mc;
        out[(size_t)NPTS * LB + (size_t)n * LB + l]   = os - cv + ss;
    }
}

extern "C" void kernel_launch(void* const* d_in, const int* in_sizes, int n_in,
                              void* d_out, int out_size, void* d_ws, size_t ws_size,
                              hipStream_t stream) {
    (void)n_in; (void)out_size; (void)d_ws; (void)ws_size;
    const float* x      = (const float*)d_in[0];
    const float* X_near = (const float*)d_in[1];
    const float* Z_near = (const float*)d_in[2];
    const float* std_Z  = (const float*)d_in[3];
    const float* ls     = (const float*)d_in[4];
    const float* os     = (const float*)d_in[5];
    const float* mc     = (const float*)d_in[6];
    float* out = (float*)d_out;

    const int n = in_sizes[0] / DD;            // 2048
    const int blocks = n / WAVES_PER_BLOCK;    // 1024
    gp_posterior_kernel<<<blocks, WAVES_PER_BLOCK * 32, 0, stream>>>(
        x, X_near, Z_near, std_Z, ls, os, mc, out);
}